// TemporalAttentionLayer_recovery_69114613727325
// MI455X (gfx1250) — compile-verified
//
#include <hip/hip_runtime.h>
#include <math.h>

// ---------------------------------------------------------------------------
// Fused temporal attention for MI455X (gfx1250, wave32, WMMA).
// One batch element per 128-thread workgroup (4 waves).
//   Phase 1: stage kv = concat(nb_feat, nb_time) (20x272) into LDS.
//   Phase 2: K = kv @ Wk^T + bk, V = kv @ Wv^T + bv via V_WMMA_F32_16X16X4_F32.
//   Phase 3: q matvec, masked softmax over 20 neighbors, weighted-V, out proj.
// All in one pass: HBM traffic == input size (~3.0 GB) -> ~128 us floor at
// 23.3 TB/s. Weights (~300 KB) are L2-resident and broadcast.
// ---------------------------------------------------------------------------

typedef __attribute__((ext_vector_type(2))) float v2f;
typedef __attribute__((ext_vector_type(8))) float v8f;

#define BN      20      // neighbors
#define FEATD   172
#define TIMED   100
#define KDIM    272     // FEAT + TIME
#define EDIM    100
#define DHEAD   50
#define MROWS   32      // 20 rows padded to 2 WMMA M-tiles
#define NTILES  7       // 100 cols padded to 112 (7 x 16)
#define KV_STRIDE  274  // 274 % 64 == 18 -> conflict-friendly LDS stride
#define OUT_STRIDE 114

__global__ __launch_bounds__(128) void temporal_attn_fused(
    const float* __restrict__ src_time,        // [B,1,100]
    const float* __restrict__ nb_feat,         // [B,20,172]
    const float* __restrict__ nb_time,         // [B,20,100]
    const unsigned char* __restrict__ pad_mask,// [B,20] (numpy bool = 1 byte)
    const float* __restrict__ Wq,              // [100,100]
    const float* __restrict__ Wk,              // [100,272]
    const float* __restrict__ Wv,              // [100,272]
    const float* __restrict__ b_in,            // [300] = bq|bk|bv
    const float* __restrict__ Wo,              // [100,100]
    const float* __restrict__ bo,              // [100]
    float* __restrict__ out,                   // [B,100]
    float* __restrict__ w_avg)                 // [B,20]
{
    __shared__ float lds_kv[MROWS * KV_STRIDE];   // 35.1 KB
    __shared__ float lds_k [BN * OUT_STRIDE];     //  9.1 KB
    __shared__ float lds_v [BN * OUT_STRIDE];     //  9.1 KB
    __shared__ float lds_q [EDIM];
    __shared__ float lds_s [2 * BN];
    __shared__ float lds_w [2 * BN];
    __shared__ float lds_o [EDIM];
    __shared__ int   lds_invalid;
    __shared__ unsigned char lds_mask[BN];

    const int b    = blockIdx.x;
    const int tid  = threadIdx.x;
    const int wid  = tid >> 5;
    const int lane = tid & 31;

    // ---------------- Phase 1: stage kv into LDS (coalesced) ---------------
    for (int i = tid; i < BN * KDIM; i += 128) {
        const int n = i / KDIM, c = i % KDIM;
        const float vdat = (c < FEATD)
            ? nb_feat[(size_t)b * (BN * FEATD) + n * FEATD + c]
            : nb_time[(size_t)b * (BN * TIMED) + n * TIMED + (c - FEATD)];
        lds_kv[n * KV_STRIDE + c] = vdat;
    }
    if (tid == 0) {
        int allm = 1;
        for (int n = 0; n < BN; ++n) {
            const unsigned char m = pad_mask[(size_t)b * BN + n] ? 1 : 0;
            lds_mask[n] = m;
            allm &= (int)m;
        }
        lds_invalid = allm;
        if (allm) lds_mask[0] = 0;   // reference un-masks neighbor 0
    }
    __syncthreads();

    // ---------------- Phase 2: K/V projection via fp32 WMMA ----------------
    // 28 jobs = {K,V} x 2 M-tiles x 7 N-tiles, round-robin over 4 waves.
    for (int job = wid; job < 2 * 2 * NTILES; job += 4) {
        const int mat = job & 1;          // 0 -> K, 1 -> V
        const int mt  = (job >> 1) & 1;   // M tile
        const int nt  = job >> 2;         // N tile
        const float* W   = mat ? Wv : Wk;                  // [100][272] row-major
        const float* bb  = b_in + (mat ? 2 * EDIM : EDIM); // bv : bk
        float* dst       = mat ? lds_v : lds_k;

        const int m0   = mt * 16;
        const int n0   = nt * 16;
        const int arow = m0 + (lane & 15);        // A: row m, K pair per half-wave
        const int koff = (lane >> 4) * 2;         // lanes 0-15: K+0/1, 16-31: K+2/3
        const int bcol = n0 + (lane & 15);        // B/C: col n = lane%16
        const int bcolc = (bcol < EDIM) ? bcol : (EDIM - 1); // clamp (junk cols unstored)

        const float* ap = lds_kv + arow * KV_STRIDE + koff;
        const float* bp = W + (size_t)bcolc * KDIM + koff;

        v8f c = {0.f, 0.f, 0.f, 0.f, 0.f, 0.f, 0.f, 0.f};
        for (int k0 = 0; k0 < KDIM; k0 += 4) {
            v2f a;  a.x  = ap[k0];  a.y  = ap[k0 + 1];   // A 16x4 fragment
            v2f bf; bf.x = bp[k0];  bf.y = bp[k0 + 1];   // B 4x16 fragment (W^T)
            c = __builtin_amdgcn_wmma_f32_16x16x4_f32(
                    /*neg_a=*/false, a, /*neg_b=*/false, bf,
                    /*c_mod=*/(short)0, c, /*reuse_a=*/false, /*reuse_b=*/false);
        }
        if (bcol < EDIM) {
            const float bias_n = bb[bcol];
            const int rbase = m0 + ((lane >> 4) << 3); // C: VGPR i -> M=i / i+8
            #pragma unroll
            for (int i = 0; i < 8; ++i) {
                const int r = rbase + i;
                if (r < BN) dst[r * OUT_STRIDE + bcol] = c[i] + bias_n;
            }
        }
    }

    // q projection overlaps tail of GEMM (touches only globals + lds_q)
    if (tid < EDIM) {
        const float* s  = src_time + (size_t)b * TIMED;
        const float* wr = Wq + (size_t)tid * EDIM;
        float acc = b_in[tid];                    // bq
        for (int t = 0; t < TIMED; ++t) acc = fmaf(s[t], wr[t], acc);
        lds_q[tid] = acc;
    }
    __syncthreads();

    // ---------------- Phase 3: attention (tiny, VALU) ----------------------
    if (tid < 2 * BN) {                            // 40 (head, neighbor) scores
        const int h = tid / BN, n = tid % BN;
        const float* krow = lds_k + n * OUT_STRIDE + h * DHEAD;
        const float* qh   = lds_q + h * DHEAD;
        float sacc = 0.f;
        for (int j = 0; j < DHEAD; ++j) sacc = fmaf(qh[j], krow[j], sacc);
        sacc *= 0.14142135623730951f;              // 1/sqrt(d), d = 50
        lds_s[tid] = lds_mask[n] ? -INFINITY : sacc;
    }
    __syncthreads();

    if (tid < 2) {                                 // softmax per head (20 wide)
        const float* sp = lds_s + tid * BN;
        float* wp = lds_w + tid * BN;
        float m = -INFINITY;
        for (int n = 0; n < BN; ++n) m = fmaxf(m, sp[n]);
        float sum = 0.f;
        for (int n = 0; n < BN; ++n) { const float e = __expf(sp[n] - m); wp[n] = e; sum += e; }
        const float inv = 1.f / sum;
        for (int n = 0; n < BN; ++n) wp[n] *= inv;
    }
    __syncthreads();

    const int invalid = lds_invalid;
    if (tid < BN) {
        w_avg[(size_t)b * BN + tid] =
            invalid ? 0.f : 0.5f * (lds_w[tid] + lds_w[BN + tid]);
    }
    if (tid < EDIM) {                              // out_h = sum_n w[h][n] * V[n][e]
        const int h = tid / DHEAD;
        const float* wp = lds_w + h * BN;
        float acc = 0.f;
        for (int n = 0; n < BN; ++n) acc = fmaf(wp[n], lds_v[n * OUT_STRIDE + tid], acc);
        lds_o[tid] = acc;
    }
    __syncthreads();

    if (tid < EDIM) {                              // out = o @ Wo^T + bo
        const float* wr = Wo + (size_t)tid * EDIM;
        float acc = bo[tid];
        for (int t = 0; t < EDIM; ++t) acc = fmaf(lds_o[t], wr[t], acc);
        out[(size_t)b * EDIM + tid] = invalid ? 0.f : acc;
    }
}

extern "C" void kernel_launch(void* const* d_in, const int* in_sizes, int n_in,
                              void* d_out, int out_size, void* d_ws, size_t ws_size,
                              hipStream_t stream) {
    (void)n_in; (void)out_size; (void)d_ws; (void)ws_size;
    const float* src_time = (const float*)d_in[0];
    const float* nb_feat  = (const float*)d_in[1];
    const float* nb_time  = (const float*)d_in[2];
    const unsigned char* pad_mask = (const unsigned char*)d_in[3]; // numpy bool
    const float* Wq   = (const float*)d_in[4];
    const float* Wk   = (const float*)d_in[5];
    const float* Wv   = (const float*)d_in[6];
    const float* b_in = (const float*)d_in[7];
    const float* Wo   = (const float*)d_in[8];
    const float* bo   = (const float*)d_in[9];

    const int B = in_sizes[3] / BN;                // mask is [B,20]
    float* out   = (float*)d_out;                  // tuple flat: [B*100] then [B*20]
    float* w_avg = out + (size_t)B * EDIM;

    temporal_attn_fused<<<B, 128, 0, stream>>>(
        src_time, nb_feat, nb_time, pad_mask,
        Wq, Wk, Wv, b_in, Wo, bo, out, w_avg);
}